// GraphSAGEBlock_1365799600616
// MI455X (gfx1250) — compile-verified
//
#include <hip/hip_runtime.h>

#define B_   16
#define C_   256
#define N_   1024
#define K_   9
#define NT_  (B_ * N_)        // 16384 tokens total
#define EPS_ 1e-5f

typedef float v2f __attribute__((ext_vector_type(2)));
typedef float v8f __attribute__((ext_vector_type(8)));
typedef int   gvec4 __attribute__((vector_size(4 * sizeof(int))));  // matches builtin param

#if defined(__has_builtin)
#  if __has_builtin(__builtin_amdgcn_global_load_async_to_lds_b128)
#    define HAVE_ASYNC_LDS 1
#  endif
#  if __has_builtin(__builtin_amdgcn_s_wait_asynccnt)
#    define HAVE_WAIT_ASYNC 1
#  endif
#endif

// ---------------------------------------------------------------------------
// CDNA5 f32 WMMA: D(16x16) = A(16x4) * B(4x16) + C
// 8-arg form: (neg_a, A, neg_b, B, c_mod, C, reuse_a, reuse_b)
// ---------------------------------------------------------------------------
__device__ __forceinline__ v8f wmma_f32(v2f a, v2f b, v8f c) {
  return __builtin_amdgcn_wmma_f32_16x16x4_f32(false, a, false, b, (short)0, c,
                                               false, false);
}

// Row-major contiguous-K operand tile loader.
// A-tile:  A[m][kk]  = base[(r0+m)*stride + k+kk]   (lanes index m)
// B-tile:  B[kk][n]  = base[(r0+n)*stride + k+kk]   (lanes index n)
// Per ISA layout: lane&15 selects the row/col; lane>=16 half holds K+2,K+3.
__device__ __forceinline__ v2f tile_rm(const float* base, int stride, int r0, int k) {
  const int lane = threadIdx.x & 31;
  const float* p = base + (size_t)(r0 + (lane & 15)) * stride + k + ((lane >> 4) << 1);
  v2f r; r.x = p[0]; r.y = p[1];
  return r;
}

// Stage a 16x256 f32 tile (16 KB) global -> LDS with the CDNA5 async copy
// path when available (ASYNCcnt-tracked b128 transfers), else plain stores.
__device__ __forceinline__ void stage_16x256(const float* gsrc, float* ldst, int tid) {
#if HAVE_ASYNC_LDS
  #pragma unroll
  for (int i = 0; i < 4; ++i) {
    const size_t off = (size_t)(i * 256 + tid) * 4;   // float4 granules
    __builtin_amdgcn_global_load_async_to_lds_b128(
        (gvec4*)(gsrc + off), (gvec4*)(ldst + off), 0, 0);
  }
#else
  for (int i = tid; i < 16 * C_; i += 256) ldst[i] = gsrc[i];
#endif
}

__device__ __forceinline__ void stage_wait() {
#if HAVE_ASYNC_LDS
#  if HAVE_WAIT_ASYNC
  __builtin_amdgcn_s_wait_asynccnt(0);
#  else
  asm volatile("s_wait_asynccnt 0" ::: "memory");
#  endif
#endif
}

// ---------------------------------------------------------------------------
// 1) transpose x[B,C,N] -> tok[B,N,C]
// ---------------------------------------------------------------------------
__global__ __launch_bounds__(256) void k_transpose(const float* __restrict__ x,
                                                   float* __restrict__ tok) {
  __shared__ float t[32][33];
  const int b  = blockIdx.z;
  const int n0 = blockIdx.x * 32;
  const int c0 = blockIdx.y * 32;
  const float* xb = x + (size_t)b * C_ * N_;
  float* tb = tok + (size_t)b * N_ * C_;
  for (int i = threadIdx.y; i < 32; i += 8)
    t[i][threadIdx.x] = xb[(size_t)(c0 + i) * N_ + n0 + threadIdx.x];
  __syncthreads();
  for (int i = threadIdx.y; i < 32; i += 8)
    tb[(size_t)(n0 + i) * C_ + c0 + threadIdx.x] = t[threadIdx.x][i];
}

// ---------------------------------------------------------------------------
// 2) squared norms sq[t] = sum_c tok[t][c]^2  (one 256-thread block per token)
// ---------------------------------------------------------------------------
__global__ __launch_bounds__(256) void k_sqnorm(const float* __restrict__ tok,
                                                float* __restrict__ sq) {
  const int t = blockIdx.x;
  float v = tok[(size_t)t * C_ + threadIdx.x];
  v *= v;
  for (int o = 16; o > 0; o >>= 1) v += __shfl_down(v, o, 32);
  __shared__ float part[8];
  if ((threadIdx.x & 31) == 0) part[threadIdx.x >> 5] = v;
  __syncthreads();
  if (threadIdx.x == 0) {
    float s = 0.f;
    #pragma unroll
    for (int i = 0; i < 8; ++i) s += part[i];
    sq[t] = s;
  }
}

// ---------------------------------------------------------------------------
// 3) fuse weights: Wa = Wcomb[:, :C] @ W_self,  Wb = Wcomb[:, C:] @ W_nb
//    one wave per 16x16 output tile; WMMA K-loop over j.
// ---------------------------------------------------------------------------
__global__ __launch_bounds__(32) void k_fuse_weights(const float* __restrict__ Wcomb,
                                                     const float* __restrict__ Wself,
                                                     const float* __restrict__ Wnb,
                                                     float* __restrict__ Wa,
                                                     float* __restrict__ Wb) {
  const int id   = blockIdx.x;
  const int sel  = id & 1;           // 0 -> Wa, 1 -> Wb
  const int tile = id >> 1;          // 0..255
  const int c0   = (tile >> 4) << 4; // output row tile (c)
  const int k0   = (tile & 15) << 4; // output col tile (k)
  const float* Win  = sel ? Wnb : Wself;
  const float* Acol = Wcomb + (sel ? C_ : 0);   // row stride 2C
  float* out = sel ? Wb : Wa;
  const int lane = threadIdx.x;

  v8f acc = 0.0f;
  for (int j = 0; j < C_; j += 4) {
    v2f a = tile_rm(Acol, 2 * C_, c0, j);       // A[m][kk] = Wcomb[c0+m][j+kk]
    // B[kk][n] = Win[j+kk][k0+n]  (natural orientation: K indexes rows)
    const float* bp = Win + (size_t)(j + ((lane >> 4) << 1)) * C_ + k0 + (lane & 15);
    v2f b; b.x = bp[0]; b.y = bp[C_];
    acc = wmma_f32(a, b, acc);
  }
  const int rbase = (lane >> 4) << 3;
  #pragma unroll
  for (int r = 0; r < 8; ++r)
    out[(size_t)(c0 + rbase + r) * C_ + k0 + (lane & 15)] = acc[r];
}

// bias_f[c] = b_comb[c] + sum_j Wcomb[c][j]*b_self[j] + Wcomb[c][C+j]*b_nb[j]
__global__ __launch_bounds__(256) void k_fuse_bias(const float* __restrict__ Wcomb,
                                                   const float* __restrict__ bself,
                                                   const float* __restrict__ bnb,
                                                   const float* __restrict__ bcomb,
                                                   float* __restrict__ biasf) {
  const int c = threadIdx.x;
  const float* row = Wcomb + (size_t)c * 2 * C_;
  float s = bcomb[c];
  for (int j = 0; j < C_; ++j) s += row[j] * bself[j] + row[C_ + j] * bnb[j];
  biasf[c] = s;
}

// ---------------------------------------------------------------------------
// 4) distances (WMMA Gram) + top-9 + neighbor mean.
//    One WG = 16 rows of one batch. 8 waves tile all 1024 columns in two
//    512-column halves (d2 half-tile lives in LDS), 16 threads keep register
//    top-9 lists, then 256 threads average the 9 neighbor rows per row.
// ---------------------------------------------------------------------------
__global__ __launch_bounds__(256) void k_dist_topk_nbmean(const float* __restrict__ tok,
                                                          const float* __restrict__ sq,
                                                          float* __restrict__ nbm) {
  __shared__ float ldsA[16 * C_];     // 16 KB: staged A rows (async path)
  __shared__ float ldsD[16 * 512];    // 32 KB: d2 half-tile
  __shared__ float ldsSqr[16];
  __shared__ int   ldsI[16][K_];

  const int wg = blockIdx.x;          // 0..1023
  const int b  = wg >> 6;
  const int r0 = (wg & 63) << 4;
  const float* tokb = tok + (size_t)b * N_ * C_;
  const float* sqb  = sq + (size_t)b * N_;
  const int tid  = threadIdx.x;
  const int wave = tid >> 5;
  const int lane = tid & 31;

  stage_16x256(tokb + (size_t)r0 * C_, ldsA, tid);
  if (tid < 16) ldsSqr[tid] = sqb[r0 + tid];
  stage_wait();
  __syncthreads();

  float best[K_];
  int   besti[K_];
  #pragma unroll
  for (int j = 0; j < K_; ++j) { best[j] = 3.4e38f; besti[j] = 0; }

  for (int half = 0; half < 2; ++half) {
    // each wave computes 4 of the 32 column tiles in this half
    for (int t = 0; t < 4; ++t) {
      const int ct = half * 32 + wave * 4 + t;   // global col tile 0..63
      const int c0 = ct << 4;
      // prefetch next tile's B rows into cache while this tile computes
      const int cn = c0 + 16;
      if (cn < N_)
        __builtin_prefetch(tokb + (size_t)(cn + (lane & 15)) * C_, 0, 0);
      v8f acc = 0.0f;
      for (int k = 0; k < C_; k += 4) {
        v2f a = tile_rm(ldsA, C_, 0, k);          // A: 16 staged rows
        v2f bb = tile_rm(tokb, C_, c0, k);        // B[kk][n] = tok[c0+n][k+kk]
        acc = wmma_f32(a, bb, acc);
      }
      const int colL = (c0 - half * 512) + (lane & 15);
      const float sqc = sqb[c0 + (lane & 15)];
      const int rbase = (lane >> 4) << 3;
      #pragma unroll
      for (int r = 0; r < 8; ++r) {
        const int row = rbase + r;
        ldsD[row * 512 + colL] = ldsSqr[row] + sqc - 2.0f * acc[r];
      }
    }
    __syncthreads();
    if (tid < 16) {
      const int row = tid;
      for (int cL = 0; cL < 512; ++cL) {
        const float d = ldsD[row * 512 + cL];
        if (d < best[K_ - 1]) {
          best[K_ - 1]  = d;
          besti[K_ - 1] = half * 512 + cL;
          #pragma unroll
          for (int j = K_ - 1; j > 0; --j) {
            if (best[j] < best[j - 1]) {
              float tf = best[j];  best[j]  = best[j - 1];  best[j - 1]  = tf;
              int   ti = besti[j]; besti[j] = besti[j - 1]; besti[j - 1] = ti;
            }
          }
        }
      }
    }
    __syncthreads();
  }

  if (tid < 16) {
    #pragma unroll
    for (int j = 0; j < K_; ++j) ldsI[tid][j] = besti[j];
  }
  __syncthreads();

  // neighbor mean: thread = channel
  for (int r = 0; r < 16; ++r) {
    float s = 0.f;
    #pragma unroll
    for (int j = 0; j < K_; ++j) s += tokb[(size_t)ldsI[r][j] * C_ + tid];
    nbm[(size_t)(b * N_ + r0 + r) * C_ + tid] = s * (1.0f / 9.0f);
  }
}

// ---------------------------------------------------------------------------
// 5) out_lin[t][c] = sum_k tok[t][k]*Wa[c][k] + sum_k nbm[t][k]*Wb[c][k] + biasf[c]
//    One WG = 16 token rows; A operands staged in LDS, 8 waves x 2 col tiles.
// ---------------------------------------------------------------------------
__global__ __launch_bounds__(256) void k_linear(const float* __restrict__ tok,
                                                const float* __restrict__ nbm,
                                                const float* __restrict__ Wa,
                                                const float* __restrict__ Wb,
                                                const float* __restrict__ biasf,
                                                float* __restrict__ outlin) {
  __shared__ float ldsA[16 * C_];
  __shared__ float ldsN[16 * C_];
  const int n0   = blockIdx.x << 4;   // global token row tile
  const int tid  = threadIdx.x;
  const int wave = tid >> 5;
  const int lane = tid & 31;

  stage_16x256(tok + (size_t)n0 * C_, ldsA, tid);
  stage_16x256(nbm + (size_t)n0 * C_, ldsN, tid);
  stage_wait();
  __syncthreads();

  for (int t = 0; t < 2; ++t) {
    const int c0 = (wave * 2 + t) << 4;
    v8f acc = 0.0f;
    for (int k = 0; k < C_; k += 4) {
      v2f a  = tile_rm(ldsA, C_, 0, k);
      v2f bb = tile_rm(Wa, C_, c0, k);   // B[kk][n] = Wa[c0+n][k+kk]
      acc = wmma_f32(a, bb, acc);
    }
    for (int k = 0; k < C_; k += 4) {
      v2f a  = tile_rm(ldsN, C_, 0, k);
      v2f bb = tile_rm(Wb, C_, c0, k);
      acc = wmma_f32(a, bb, acc);
    }
    const int col = c0 + (lane & 15);
    const float bf = biasf[col];
    const int rbase = (lane >> 4) << 3;
    #pragma unroll
    for (int r = 0; r < 8; ++r)
      outlin[(size_t)(n0 + rbase + r) * C_ + col] = acc[r] + bf;
  }
}

// ---------------------------------------------------------------------------
// 6) BatchNorm stats + finalize + elementwise epilogue
// ---------------------------------------------------------------------------
__global__ __launch_bounds__(512) void k_bn_zero(float* __restrict__ sums,
                                                 float* __restrict__ sumsq) {
  if (threadIdx.x < C_) sums[threadIdx.x] = 0.f;
  else                  sumsq[threadIdx.x - C_] = 0.f;
}

__global__ __launch_bounds__(256) void k_bn_partial(const float* __restrict__ outlin,
                                                    float* __restrict__ sums,
                                                    float* __restrict__ sumsq) {
  const int c  = threadIdx.x;
  const int i0 = blockIdx.x * 256;
  float s = 0.f, s2 = 0.f;
  for (int i = 0; i < 256; ++i) {
    const float v = outlin[(size_t)(i0 + i) * C_ + c];
    s += v; s2 += v * v;
  }
  atomicAdd(&sums[c], s);
  atomicAdd(&sumsq[c], s2);
}

__global__ __launch_bounds__(256) void k_bn_finalize(const float* __restrict__ sums,
                                                     const float* __restrict__ sumsq,
                                                     const float* __restrict__ gamma,
                                                     const float* __restrict__ beta,
                                                     float* __restrict__ scale,
                                                     float* __restrict__ shift) {
  const int c = threadIdx.x;
  const float inv = 1.0f / (float)NT_;
  const float mu  = sums[c] * inv;
  const float var = sumsq[c] * inv - mu * mu;
  const float rs  = rsqrtf(var + EPS_);
  const float sc  = gamma[c] * rs;
  scale[c] = sc;
  shift[c] = beta[c] - mu * sc;
}

__global__ __launch_bounds__(256) void k_epilogue(const float* __restrict__ outlin,
                                                  const float* __restrict__ scale,
                                                  const float* __restrict__ shift,
                                                  const float* __restrict__ x,
                                                  float* __restrict__ out) {
  const size_t gid = (size_t)blockIdx.x * 256 + threadIdx.x;  // [B][C][N] layout
  const int n = (int)(gid & (N_ - 1));
  const int c = (int)((gid >> 10) & (C_ - 1));
  const int b = (int)(gid >> 18);
  const float v = outlin[(size_t)(b * N_ + n) * C_ + c] * scale[c] + shift[c] + x[gid];
  out[gid] = fmaxf(v, 0.f);
}

// ---------------------------------------------------------------------------
extern "C" void kernel_launch(void* const* d_in, const int* in_sizes, int n_in,
                              void* d_out, int out_size, void* d_ws, size_t ws_size,
                              hipStream_t stream) {
  const float* x      = (const float*)d_in[0];
  const float* Wself  = (const float*)d_in[1];
  const float* bself  = (const float*)d_in[2];
  const float* Wnb    = (const float*)d_in[3];
  const float* bnb    = (const float*)d_in[4];
  const float* Wcomb  = (const float*)d_in[5];
  const float* bcomb  = (const float*)d_in[6];
  const float* gamma  = (const float*)d_in[7];
  const float* beta   = (const float*)d_in[8];
  float* out = (float*)d_out;

  float* W = (float*)d_ws;
  float* tok    = W;                      // 16384*256
  float* nbm    = tok    + (size_t)NT_ * C_;
  float* outlin = nbm    + (size_t)NT_ * C_;
  float* Wa     = outlin + (size_t)NT_ * C_;
  float* Wb     = Wa     + (size_t)C_ * C_;
  float* biasf  = Wb     + (size_t)C_ * C_;
  float* sq     = biasf  + C_;            // 16384
  float* sums   = sq     + NT_;
  float* sumsq  = sums   + C_;
  float* scale  = sumsq  + C_;
  float* shift  = scale  + C_;

  k_transpose<<<dim3(N_ / 32, C_ / 32, B_), dim3(32, 8), 0, stream>>>(x, tok);
  k_sqnorm<<<NT_, 256, 0, stream>>>(tok, sq);
  k_fuse_weights<<<512, 32, 0, stream>>>(Wcomb, Wself, Wnb, Wa, Wb);
  k_fuse_bias<<<1, 256, 0, stream>>>(Wcomb, bself, bnb, bcomb, biasf);
  k_dist_topk_nbmean<<<B_ * (N_ / 16), 256, 0, stream>>>(tok, sq, nbm);
  k_linear<<<NT_ / 16, 256, 0, stream>>>(tok, nbm, Wa, Wb, biasf, outlin);
  k_bn_zero<<<1, 512, 0, stream>>>(sums, sumsq);
  k_bn_partial<<<NT_ / 256, 256, 0, stream>>>(outlin, sums, sumsq);
  k_bn_finalize<<<1, 256, 0, stream>>>(sums, sumsq, gamma, beta, scale, shift);
  k_epilogue<<<(B_ * C_ * N_) / 256, 256, 0, stream>>>(outlin, scale, shift, x, out);
}